// DvD_38147899523363
// MI455X (gfx1250) — compile-verified
//
#include <hip/hip_runtime.h>
#include <stdint.h>

// ---------------------------------------------------------------------------
// det( 0.5 * (normalize_rows(A) @ normalize_rows(A)^T + 1) ),  A: 64 x 524288 f32
//
//   1) zero 64x64 accumulator in workspace
//   2) Gram = A @ A^T via V_WMMA_F32_16X16X4_F32; K-chunks staged into a
//      double-buffered LDS slab by the Tensor Data Mover (tensor_load_to_lds,
//      TENSORcnt-tracked, wave0-issued), overlapping DMA with WMMA compute.
//   3) one small workgroup: normalize by diag, 0.5*(.+1), LU det (PD, no pivot)
// ---------------------------------------------------------------------------

typedef float    v2f __attribute__((ext_vector_type(2)));
typedef float    v8f __attribute__((ext_vector_type(8)));
typedef uint32_t su4 __attribute__((ext_vector_type(4)));
typedef uint32_t su8 __attribute__((ext_vector_type(8)));

#define N_ROWS     64
#define CHUNK      256            // K-columns per LDS slab
#define LDS_STRIDE 260            // 256 + 4 dword pad -> conflict-free frag reads
#define SLAB       (N_ROWS * LDS_STRIDE)   // floats per buffer
#define WG_THREADS 512            // 16 waves (wave32): one 16x16 tile per wave
#define NUM_WG     128

__global__ void DvD_zero_gram(float* __restrict__ gram) {
    int t = threadIdx.x + blockIdx.x * blockDim.x;
    if (t < N_ROWS * N_ROWS) gram[t] = 0.0f;
}

// Issue one TDM 2D tile load: 256 x 64 f32 tile, row stride Dtot elements,
// LDS destination padded by 4 dwords every 256 dwords (-> 260-dword rows).
// D# packing per CDNA5 ISA 8.3 (group0) / 8.4 (group1); 2D => groups 2/3 NULL.
__device__ __forceinline__ void tdm_load_tile(const float* A, int Dtot,
                                              long base, uint32_t ldsOff) {
    uint64_t gaddr = (uint64_t)(uintptr_t)A + (uint64_t)base * 4u;

    su4 g0;
    g0[0] = 1u;                                          // count=1, user D#
    g0[1] = ldsOff;                                      // lds_addr (bytes)
    g0[2] = (uint32_t)gaddr;                             // global_addr[31:0]
    g0[3] = (uint32_t)((gaddr >> 32) & 0x01FFFFFFull)    // global_addr[56:32]
          | (2u << 30);                                  // type=2 ("image")

    uint32_t td0     = (uint32_t)(Dtot - base);          // elems left in row (OOB->0 fill)
    uint32_t td1     = N_ROWS;                           // 64 rows
    uint64_t stride0 = (uint64_t)Dtot;                   // row stride, elements

    su8 g1;
    g1[0] = (2u << 16)                                   // data_size = 4B
          | (1u << 20)                                   // pad_enable
          | (7u << 22)                                   // pad_interval: 256 dwords
          | (3u << 25);                                  // pad_amount: 4 dwords
    g1[1] = (td0 & 0xFFFFu) << 16;                       // tensor_dim0[15:0] @bit48
    g1[2] = (td0 >> 16) | ((td1 & 0xFFFFu) << 16);       // tdim0 hi / tdim1 lo
    g1[3] = (td1 >> 16) | ((uint32_t)CHUNK << 16);       // tile_dim0 = 256
    g1[4] = (uint32_t)N_ROWS;                            // tile_dim1 = 64, tile_dim2 = 0
    g1[5] = (uint32_t)stride0;                           // tensor_dim0_stride lo32
    g1[6] = (uint32_t)(stride0 >> 32) & 0xFFFFu;         // stride0 hi16; stride1 lo = 0
    g1[7] = 0u;                                          // stride1 hi = 0

    asm volatile("tensor_load_to_lds %0, %1" :: "s"(g0), "s"(g1) : "memory");
}

__global__ __launch_bounds__(WG_THREADS)
void DvD_gram_wmma(const float* __restrict__ A, float* __restrict__ gram,
                   int Dtot, int nchunks) {
    __shared__ float lds[2 * SLAB];

    const int tid  = threadIdx.x;
    const int lane = tid & 31;           // wave32
    const int wave = tid >> 5;           // 0..15
    const int ti   = wave >> 2;          // output row-tile (0..3)
    const int tj   = wave & 3;           // output col-tile (0..3)

    // fp32 A/B fragment layout (ISA 7.12.2, 16x4 f32):
    //   lanes 0-15 : VGPR0=K+0, VGPR1=K+1   (M/N = lane)
    //   lanes16-31 : VGPR0=K+2, VGPR1=K+3   (M/N = lane-16)
    const int frow = lane & 15;
    const int koff = (lane >> 4) << 1;

    // AS3 offsets of the two slabs (low 32 bits of generic LDS pointer).
    const uint32_t off0 = (uint32_t)(uintptr_t)(&lds[0]);
    const uint32_t off1 = (uint32_t)(uintptr_t)(&lds[SLAB]);

    const int c0 = blockIdx.x;           // grid is clamped to <= nchunks
    if (wave == 0)
        tdm_load_tile(A, Dtot, (long)c0 * CHUNK, off0);  // prologue DMA -> buf0

    v8f acc = {};                        // persistent fp32 accumulator tile
    int p = 0;
    for (int chunk = c0; chunk < nchunks; chunk += gridDim.x, ++p) {
        const float*   cur    = (p & 1) ? (lds + SLAB) : lds;
        const uint32_t nxtOff = (p & 1) ? off0 : off1;
        const int      next   = chunk + gridDim.x;

        if (wave == 0) {
            if (next < nchunks) {
                tdm_load_tile(A, Dtot, (long)next * CHUNK, nxtOff); // prefetch
                __builtin_amdgcn_s_wait_tensorcnt(1);  // in-order => cur resident
            } else {
                __builtin_amdgcn_s_wait_tensorcnt(0);
            }
        }
        __syncthreads();                 // cur visible to all waves

        // ---- 16x16 tile over this K-chunk: 64 x v_wmma_f32_16x16x4_f32 ----
        const float* arow = cur + (ti * 16 + frow) * LDS_STRIDE + koff;
        const float* brow = cur + (tj * 16 + frow) * LDS_STRIDE + koff;
        #pragma unroll 8
        for (int k = 0; k < CHUNK; k += 4) {
            v2f a = *(const v2f*)(arow + k);   // ds_load_2addr_b64, conflict-free
            v2f b = *(const v2f*)(brow + k);
            acc = __builtin_amdgcn_wmma_f32_16x16x4_f32(
                false, a, false, b, (short)0, acc, false, false);
        }
        __syncthreads();                 // cur free before it becomes next DMA target
    }

    // ---- fold partial tile into global gram (C/D layout: VGPR g -> M=g+8*hi) ----
    const int n   = lane & 15;
    const int mhi = (lane >> 4) << 3;
    #pragma unroll
    for (int g = 0; g < 8; ++g) {
        int m = g + mhi;
        unsafeAtomicAdd(&gram[(ti * 16 + m) * N_ROWS + (tj * 16 + n)], acc[g]);
    }
}

__global__ __launch_bounds__(64)
void DvD_det64(const float* __restrict__ gram, float* __restrict__ out) {
    __shared__ float M[64][65];
    __shared__ float nrm[64];
    __shared__ float det;

    const int r = threadIdx.x;           // one thread per row, 2 waves
    nrm[r] = sqrtf(gram[r * 64 + r]);
    if (r == 0) det = 1.0f;
    __syncthreads();

    const float inr = nrm[r];
    for (int j = 0; j < 64; ++j)
        M[r][j] = 0.5f * (gram[r * 64 + j] / (inr * nrm[j]) + 1.0f);
    __syncthreads();

    // LU without pivoting (Gram of unit vectors -> PD); det = prod(pivots)
    for (int k = 0; k < 64; ++k) {
        if (r == k) det *= M[k][k];
        __syncthreads();
        if (r > k) {
            float f = M[r][k] / M[k][k];
            for (int j = k; j < 64; ++j)
                M[r][j] -= f * M[k][j];
        }
        __syncthreads();
    }
    if (r == 0) out[0] = det;
}

extern "C" void kernel_launch(void* const* d_in, const int* in_sizes, int n_in,
                              void* d_out, int out_size, void* d_ws, size_t ws_size,
                              hipStream_t stream) {
    const float* A  = (const float*)d_in[0];
    float* gram     = (float*)d_ws;              // 64*64*4 = 16 KiB scratch
    float* out      = (float*)d_out;

    const int total   = in_sizes[0];
    const int Dtot    = total / N_ROWS;          // 524288
    const int nchunks = (Dtot + CHUNK - 1) / CHUNK;

    DvD_zero_gram<<<16, 256, 0, stream>>>(gram);

    int grid = NUM_WG < nchunks ? NUM_WG : nchunks;
    DvD_gram_wmma<<<grid, WG_THREADS, 0, stream>>>(A, gram, Dtot, nchunks);

    DvD_det64<<<1, 64, 0, stream>>>(gram, out);
}